// MatchingModelRobust_34153579938508
// MI455X (gfx1250) — compile-verified
//
#include <hip/hip_runtime.h>
#include <stdint.h>
#include <math.h>

#define B  256
#define LX 512
#define LY 256
#define SS 128
#define AA 128
#define WAVES 4

typedef __attribute__((ext_vector_type(4))) unsigned int v4u;
typedef __attribute__((ext_vector_type(8))) int v8i;
typedef __attribute__((ext_vector_type(4))) int v4i;

// ---------------- Kernel 1: cij = 1 - softmax(P, axis=1) ----------------
// One wave per row; lane handles 4 strided columns; shfl_xor reductions.
__global__ __launch_bounds__(32) void softmax_kernel(const float* __restrict__ P,
                                                     float* __restrict__ cij) {
  const int r = blockIdx.x;
  const int lane = threadIdx.x;
  float p[4];
#pragma unroll
  for (int t = 0; t < 4; ++t) p[t] = P[r * AA + lane + 32 * t];
  float m = fmaxf(fmaxf(p[0], p[1]), fmaxf(p[2], p[3]));
#pragma unroll
  for (int d = 16; d >= 1; d >>= 1) m = fmaxf(m, __shfl_xor(m, d));
  float e[4];
  float s = 0.f;
#pragma unroll
  for (int t = 0; t < 4; ++t) { e[t] = expf(p[t] - m); s += e[t]; }
#pragma unroll
  for (int d = 16; d >= 1; d >>= 1) s += __shfl_xor(s, d);
  float inv = 1.0f / s;
#pragma unroll
  for (int t = 0; t < 4; ++t) cij[r * AA + lane + 32 * t] = 1.0f - e[t] * inv;
}

// ---------------- Kernel 2: edit DP, one wave per batch element ----------------
__global__ __launch_bounds__(32 * WAVES) void dp_kernel(
    const int* __restrict__ xs, const int* __restrict__ ys,
    const int* __restrict__ xl, const int* __restrict__ yl,
    const int* __restrict__ dn, const float* __restrict__ cij_g,
    float* __restrict__ per_seq) {
  __shared__ float cij[SS * AA];  // 64 KB cost table, shared by 4 waves
  const int lane = threadIdx.x & 31;
  const int wave = threadIdx.x >> 5;

  // ---- Stage cij into LDS via the Tensor Data Mover (CDNA5 TDM) ----
#if __has_builtin(__builtin_amdgcn_tensor_load_to_lds) && __has_builtin(__builtin_amdgcn_s_wait_tensorcnt)
  if (wave == 0) {
    unsigned long long ga = (unsigned long long)(uintptr_t)cij_g;
    unsigned int lds_off = (unsigned int)(uintptr_t)&cij[0];  // low 32b of LDS aperture addr
    v4u g0;
    g0[0] = 1u;                                    // count=1, user descriptor
    g0[1] = lds_off;                               // lds_addr
    g0[2] = (unsigned int)(ga & 0xFFFFFFFFu);      // global_addr[31:0]
    g0[3] = (unsigned int)((ga >> 32) & 0x01FFFFFFu) | (2u << 30);  // addr[56:32] | type=2
    v8i g1;
    g1[0] = 0x20000;               // workgroup_mask=0, data_size=2 (4B)
    g1[1] = (AA & 0xFFFF) << 16;   // tensor_dim0 lo16 -> bits 63:48  (=128)
    g1[2] = (SS & 0xFFFF) << 16;   // tensor_dim0 hi=0 ; tensor_dim1 lo -> bits 95:80 (=128)
    g1[3] = (AA & 0xFFFF) << 16;   // tensor_dim1 hi=0 ; tile_dim0 -> bits 127:112 (=128)
    g1[4] = SS;                    // tile_dim1 = 128, tile_dim2 = 0
    g1[5] = AA;                    // tensor_dim0_stride[31:0] = 128 elements
    g1[6] = 0;
    g1[7] = 0;
    v4i z4 = {0, 0, 0, 0};
    v8i z8 = {0, 0, 0, 0, 0, 0, 0, 0};
    __builtin_amdgcn_tensor_load_to_lds(g0, g1, z4, z4, z8, 0);
    __builtin_amdgcn_s_wait_tensorcnt(0);
  }
  __syncthreads();
#else
  for (int idx = threadIdx.x; idx < SS * AA; idx += blockDim.x) cij[idx] = cij_g[idx];
  __syncthreads();
#endif

  const int b = blockIdx.x * WAVES + wave;
  const int xlen = xl[b];
  const int ylen = yl[b];
  int d0 = dn[0]; d0 = d0 < 0 ? 0 : (d0 > 1 ? 1 : d0);
  int d1 = dn[1]; d1 = d1 < 0 ? 0 : (d1 > 1 ? 1 : d1);
  const float del = (float)d0;
  const float ins = (float)d1;

  // Lane L owns columns j = 1 + 8L .. 8 + 8L  (j=0 column is analytic: i*del)
  int   yoff[8];
  float prev[8], jins[8];
#pragma unroll
  for (int t = 0; t < 8; ++t) {
    int j = 1 + lane * 8 + t;
    yoff[t] = ys[(size_t)b * LY + (j - 1)];
    jins[t] = (float)j * ins;
    prev[t] = jins[t];  // row 0: D[0][j] = j*ins
  }

  const int* xrow = xs + (size_t)b * LX;
  int xi = (xlen > 0) ? xrow[0] : 0;
  for (int i = 1; i <= xlen; ++i) {
    int xin = (i < xlen) ? xrow[i] : 0;  // prefetch next row index
    const float* crow = &cij[xi * AA];
    float sub[8];
#pragma unroll
    for (int t = 0; t < 8; ++t) sub[t] = crow[yoff[t]];  // 8 LDS gathers

    // prev[j-1] shift: cross-lane carry of prev[7]; lane0 gets D[i-1][0]
    float carry = __shfl_up(prev[7], 1);
    if (lane == 0) carry = (float)(i - 1) * del;

    float g[8];
    g[0] = fminf(prev[0] + del, carry + sub[0]) - jins[0];
#pragma unroll
    for (int t = 1; t < 8; ++t)
      g[t] = fminf(prev[t] + del, prev[t - 1] + sub[t]) - jins[t];

    // lane-local inclusive prefix-min
    float r[8];
    r[0] = g[0];
#pragma unroll
    for (int t = 1; t < 8; ++t) r[t] = fminf(r[t - 1], g[t]);

    // wave32 inclusive min-scan of lane totals (5 shfl_up steps)
    float v = r[7];
#pragma unroll
    for (int d = 1; d < 32; d <<= 1) {
      float o = __shfl_up(v, d);
      if (lane >= d) v = fminf(v, o);
    }
    float excl = __shfl_up(v, 1);
    float g0row = (float)i * del;  // g at j=0 this row
    float pre = (lane == 0) ? g0row : fminf(excl, g0row);
#pragma unroll
    for (int t = 0; t < 8; ++t) prev[t] = fminf(pre, r[t]) + jins[t];
    xi = xin;
  }

  // Extract D[xlen][ylen]
  float res;
  if (ylen == 0) {
    res = (float)xlen * del;
  } else {
    int tl = (ylen - 1) >> 3;
    int tt = (ylen - 1) & 7;
    float val = 0.f;
#pragma unroll
    for (int t = 0; t < 8; ++t)
      if (t == tt) val = prev[t];
    res = __shfl(val, tl);
  }
  if (lane == 0) per_seq[b] = res;
}

// ---------------- Kernel 3: deterministic fixed-order mean ----------------
__global__ void reduce_kernel(const float* __restrict__ per_seq, float* __restrict__ out) {
  if (threadIdx.x == 0 && blockIdx.x == 0) {
    float s = 0.f;
    for (int i = 0; i < B; ++i) s += per_seq[i];
    out[0] = s * (1.0f / (float)B);
  }
}

extern "C" void kernel_launch(void* const* d_in, const int* in_sizes, int n_in,
                              void* d_out, int out_size, void* d_ws, size_t ws_size,
                              hipStream_t stream) {
  (void)in_sizes; (void)n_in; (void)out_size; (void)ws_size;
  const int*   xs = (const int*)d_in[0];
  const int*   ys = (const int*)d_in[1];
  const int*   xl = (const int*)d_in[2];
  const int*   yl = (const int*)d_in[3];
  const float* P  = (const float*)d_in[4];
  const int*   dn = (const int*)d_in[5];

  float* cij_g   = (float*)d_ws;          // 64 KB
  float* per_seq = cij_g + SS * AA;       // 1 KB
  float* out     = (float*)d_out;

  softmax_kernel<<<SS, 32, 0, stream>>>(P, cij_g);
  dp_kernel<<<B / WAVES, 32 * WAVES, 0, stream>>>(xs, ys, xl, yl, dn, cij_g, per_seq);
  reduce_kernel<<<1, 32, 0, stream>>>(per_seq, out);
}